// GAT_49323404427974
// MI455X (gfx1250) — compile-verified
//
#include <hip/hip_runtime.h>
#include <hip/hip_bf16.h>
#include <math.h>

typedef __attribute__((ext_vector_type(2))) float v2f;
typedef __attribute__((ext_vector_type(8))) float v8f;

#define HEADS 4
#define CHID  64
#define HC    256   // HEADS*CHID (width of h for every layer)
#define NEG_SLOPE 0.2f
#define EPS 1e-16f

#define WMMA_F32(a, b, c)                                                     \
    __builtin_amdgcn_wmma_f32_16x16x4_f32(false, (a), false, (b), (short)0,   \
                                          (c), false, false)

// ---------------------------------------------------------------------------
// WMMA GEMM: C[M,Nc] = A[M,K] * B[K,Nc], f32.
// One wave computes a 16x64 strip: 4 N-tiles register-blocked so each A
// fragment load (global_load_b64) feeds 4 V_WMMA_F32_16X16X4_F32 issues.
// A layout (ISA 32-bit A 16x4): lanes 0-15 rows M=0..15, half-wave picks the
// K pair. B mirrored (N across lanes). C/D: VGPR r -> M=r / M=r+8 per half.
// ---------------------------------------------------------------------------
__global__ void gat_wmma_gemm(const float* __restrict__ A,
                              const float* __restrict__ B,
                              float* __restrict__ C,
                              int M, int K, int Nc) {
    const int waves_per_blk = blockDim.x >> 5;
    const int wave = threadIdx.x >> 5;
    const int lane = threadIdx.x & 31;
    const int half = lane >> 4;       // 0: K pair {k,k+1}; 1: {k+2,k+3}
    const int l    = lane & 15;

    const int tileM  = blockIdx.x * waves_per_blk + wave;
    const int nbase  = blockIdx.y * 64;          // 4 consecutive 16-wide tiles
    if (tileM * 16 >= M) return;      // uniform per wave -> EXEC stays all-1s

    const float* __restrict__ arow  = A + (size_t)(tileM * 16 + l) * K;
    const float* __restrict__ bbase = B + nbase + l;

    v8f acc0 = {}, acc1 = {}, acc2 = {}, acc3 = {};
    for (int k = 0; k < K; k += 4) {
        const int ka = k + half * 2;
        v2f a = *(const v2f*)(arow + ka);              // global_load_b64
        const float* b0 = bbase + (size_t)ka * Nc;     // row k0
        const float* b1 = b0 + Nc;                     // row k0+1
        v2f bb;
        bb.x = b0[0];  bb.y = b1[0];  acc0 = WMMA_F32(a, bb, acc0);
        bb.x = b0[16]; bb.y = b1[16]; acc1 = WMMA_F32(a, bb, acc1);
        bb.x = b0[32]; bb.y = b1[32]; acc2 = WMMA_F32(a, bb, acc2);
        bb.x = b0[48]; bb.y = b1[48]; acc3 = WMMA_F32(a, bb, acc3);
    }

    const int rowbase = tileM * 16 + half * 8;
    float* __restrict__ crow = C + nbase + l;
#pragma unroll
    for (int r = 0; r < 8; ++r) {
        float* cr = crow + (size_t)(rowbase + r) * Nc;
        cr[0]  = acc0[r];
        cr[16] = acc1[r];
        cr[32] = acc2[r];
        cr[48] = acc3[r];
    }
}

// ---------------------------------------------------------------------------
// Per-(node,head) attention coefficients: a_src = <h, att_src>, a_dst likewise
// ---------------------------------------------------------------------------
__global__ void gat_att_scores(const float* __restrict__ h,
                               const float* __restrict__ att_src,
                               const float* __restrict__ att_dst,
                               float* __restrict__ a_src,
                               float* __restrict__ a_dst, int Nn) {
    int i = blockIdx.x * blockDim.x + threadIdx.x;   // i = n*HEADS + hd
    if (i >= Nn * HEADS) return;
    int n = i >> 2, hd = i & 3;
    const float* hp = h + (size_t)n * HC + hd * CHID;
    const float* as = att_src + hd * CHID;
    const float* ad = att_dst + hd * CHID;
    float ss = 0.f, sd = 0.f;
#pragma unroll 8
    for (int c = 0; c < CHID; ++c) {
        float v = hp[c];
        ss += v * as[c];
        sd += v * ad[c];
    }
    a_src[i] = ss;
    a_dst[i] = sd;
}

// ---------------------------------------------------------------------------
// Init: m = -inf, denom = 0, out accumulator = 0
// ---------------------------------------------------------------------------
__global__ void gat_init(float* __restrict__ m, float* __restrict__ denom,
                         float* __restrict__ outb, int Nn) {
    int i = blockIdx.x * blockDim.x + threadIdx.x;
    if (i < Nn * HC) outb[i] = 0.f;
    if (i < Nn * HEADS) {
        m[i] = -INFINITY;
        denom[i] = 0.f;
    }
}

__device__ __forceinline__ void atomicMaxFloat(float* addr, float val) {
    if (val >= 0.f)
        atomicMax((int*)addr, __float_as_int(val));
    else
        atomicMin((unsigned int*)addr, __float_as_uint(val));
}

__device__ __forceinline__ void edge_ends(const int* __restrict__ ei, int e,
                                          int E, int& s, int& d) {
    if (e < E) { s = ei[e]; d = ei[E + e]; }
    else       { s = e - E; d = e - E; }     // self loops appended
}

// Pass 1: per-destination running max of leaky_relu(a_src[s]+a_dst[d])
__global__ void gat_edge_max(const int* __restrict__ ei, int E, int Nn,
                             const float* __restrict__ a_src,
                             const float* __restrict__ a_dst,
                             float* __restrict__ m) {
    int e = blockIdx.x * blockDim.x + threadIdx.x;
    if (e >= E + Nn) return;
    int s, d; edge_ends(ei, e, E, s, d);
#pragma unroll
    for (int hd = 0; hd < HEADS; ++hd) {
        float v = a_src[s * HEADS + hd] + a_dst[d * HEADS + hd];
        v = v > 0.f ? v : NEG_SLOPE * v;
        atomicMaxFloat(&m[d * HEADS + hd], v);
    }
}

// Pass 2: p = exp(e - m[dst]); denom[dst] += p; cache p per edge
__global__ void gat_edge_exp(const int* __restrict__ ei, int E, int Nn,
                             const float* __restrict__ a_src,
                             const float* __restrict__ a_dst,
                             const float* __restrict__ m,
                             float* __restrict__ denom,
                             float* __restrict__ pbuf) {
    int e = blockIdx.x * blockDim.x + threadIdx.x;
    if (e >= E + Nn) return;
    int s, d; edge_ends(ei, e, E, s, d);
#pragma unroll
    for (int hd = 0; hd < HEADS; ++hd) {
        float v = a_src[s * HEADS + hd] + a_dst[d * HEADS + hd];
        v = v > 0.f ? v : NEG_SLOPE * v;
        float p = expf(v - m[d * HEADS + hd]);
        pbuf[(size_t)e * HEADS + hd] = p;
        atomicAdd(&denom[d * HEADS + hd], p);
    }
}

// Pass 3: out[dst] += h[src] * alpha ; 64 threads (channels) per edge
__global__ void gat_edge_scatter(const int* __restrict__ ei, int E, int Nn,
                                 const float* __restrict__ h,
                                 const float* __restrict__ pbuf,
                                 const float* __restrict__ denom,
                                 float* __restrict__ outb) {
    long long t = (long long)blockIdx.x * blockDim.x + threadIdx.x;
    int c = (int)(t & (CHID - 1));
    long long e = t >> 6;
    if (e >= (long long)E + Nn) return;
    int s, d; edge_ends(ei, (int)e, E, s, d);
#pragma unroll
    for (int hd = 0; hd < HEADS; ++hd) {
        float alpha = pbuf[(size_t)e * HEADS + hd] /
                      (denom[d * HEADS + hd] + EPS);
        float msg = h[(size_t)s * HC + hd * CHID + c] * alpha;
        atomicAdd(&outb[(size_t)d * HC + hd * CHID + c], msg);
    }
}

// mean over heads + bias + relu  -> x for next layer
__global__ void gat_head_mean_relu(const float* __restrict__ outb,
                                   const float* __restrict__ bias,
                                   float* __restrict__ xout, int Nn) {
    int i = blockIdx.x * blockDim.x + threadIdx.x;
    if (i >= Nn * CHID) return;
    int n = i >> 6, c = i & (CHID - 1);
    const float* o = outb + (size_t)n * HC;
    float v = 0.25f * (o[c] + o[CHID + c] + o[2 * CHID + c] + o[3 * CHID + c]) +
              bias[c];
    xout[i] = v > 0.f ? v : 0.f;
}

// row-wise log_softmax over 64 channels
__global__ void gat_log_softmax(const float* __restrict__ xin,
                                float* __restrict__ out, int Nn) {
    int n = blockIdx.x * blockDim.x + threadIdx.x;
    if (n >= Nn) return;
    const float* xr = xin + (size_t)n * CHID;
    float mx = -INFINITY;
#pragma unroll 8
    for (int c = 0; c < CHID; ++c) mx = fmaxf(mx, xr[c]);
    float s = 0.f;
#pragma unroll 8
    for (int c = 0; c < CHID; ++c) s += expf(xr[c] - mx);
    float ls = logf(s) + mx;
#pragma unroll 8
    for (int c = 0; c < CHID; ++c) out[(size_t)n * CHID + c] = xr[c] - ls;
}

// ---------------------------------------------------------------------------
extern "C" void kernel_launch(void* const* d_in, const int* in_sizes, int n_in,
                              void* d_out, int out_size, void* d_ws, size_t ws_size,
                              hipStream_t stream) {
    const float* x  = (const float*)d_in[0];
    const int*   ei = (const int*)d_in[1];     // [2, E]: src row then dst row
    const float* W[3]  = {(const float*)d_in[2], (const float*)d_in[6],  (const float*)d_in[10]};
    const float* AS[3] = {(const float*)d_in[3], (const float*)d_in[7],  (const float*)d_in[11]};
    const float* AD[3] = {(const float*)d_in[4], (const float*)d_in[8],  (const float*)d_in[12]};
    const float* Bv[3] = {(const float*)d_in[5], (const float*)d_in[9],  (const float*)d_in[13]};

    const int Nn   = in_sizes[0] / 256;   // 50000
    const int E    = in_sizes[1] / 2;     // 800000
    const int Etot = E + Nn;

    float* ws   = (float*)d_ws;
    float* h    = ws;                          // Nn*256
    float* outb = h    + (size_t)Nn * HC;      // Nn*256
    float* xbuf = outb + (size_t)Nn * HC;      // Nn*64
    float* asrc = xbuf + (size_t)Nn * CHID;    // Nn*4
    float* adst = asrc + (size_t)Nn * HEADS;   // Nn*4
    float* mbuf = adst + (size_t)Nn * HEADS;   // Nn*4
    float* dbuf = mbuf + (size_t)Nn * HEADS;   // Nn*4
    float* pbuf = dbuf + (size_t)Nn * HEADS;   // Etot*4

    const float* xin = x;
    int K = 256;

    for (int L = 0; L < 3; ++L) {
        // init max/denom/accumulator
        {
            int tot = Nn * HC;
            gat_init<<<(tot + 255) / 256, 256, 0, stream>>>(mbuf, dbuf, outb, Nn);
        }
        // h = xin @ W[L]   (WMMA f32 16x16x4, 16x64 strip per wave)
        {
            int mtiles = (Nn + 15) / 16;               // 3125
            dim3 grid((mtiles + 3) / 4, HC / 64);      // y: 4 strips of 64 cols
            gat_wmma_gemm<<<grid, 128, 0, stream>>>(xin, W[L], h, Nn, K, HC);
        }
        // per-node attention coefficients
        {
            int tot = Nn * HEADS;
            gat_att_scores<<<(tot + 255) / 256, 256, 0, stream>>>(
                h, AS[L], AD[L], asrc, adst, Nn);
        }
        // segment softmax over edges
        gat_edge_max<<<(Etot + 255) / 256, 256, 0, stream>>>(
            ei, E, Nn, asrc, adst, mbuf);
        gat_edge_exp<<<(Etot + 255) / 256, 256, 0, stream>>>(
            ei, E, Nn, asrc, adst, mbuf, dbuf, pbuf);
        // message scatter: 64 threads per edge
        {
            long long tot = (long long)Etot * CHID;
            int blocks = (int)((tot + 255) / 256);
            gat_edge_scatter<<<blocks, 256, 0, stream>>>(
                ei, E, Nn, h, pbuf, dbuf, outb);
        }
        // head mean + bias + relu -> xbuf
        {
            int tot = Nn * CHID;
            gat_head_mean_relu<<<(tot + 255) / 256, 256, 0, stream>>>(
                outb, Bv[L], xbuf, Nn);
        }
        xin = xbuf;
        K = CHID;
    }

    gat_log_softmax<<<(Nn + 255) / 256, 256, 0, stream>>>(xbuf, (float*)d_out, Nn);
}